// SAGNN_58712202936406
// MI455X (gfx1250) — compile-verified
//
#include <hip/hip_runtime.h>
#include <cmath>

static constexpr float EPS = 0.1f;
static constexpr int DF = 128;        // feature dim (D == Dout == 128)
static constexpr int WT_STRIDE = 132; // padded LDS row stride (floats): bank-conflict-free

typedef __attribute__((ext_vector_type(2))) float v2f;
typedef __attribute__((ext_vector_type(8))) float v8f;
typedef __attribute__((ext_vector_type(4))) unsigned int u32x4;
typedef __attribute__((ext_vector_type(8))) int i32x8;
typedef __attribute__((ext_vector_type(4))) int i32x4;

// ---------------------------------------------------------------------------
// Kernel 1: one wave32 per node. al = x.att_l, ar = x.att_r (float4 loads +
// shfl_xor butterfly), zero agg row, deg = 1 (self loop).
// ---------------------------------------------------------------------------
__global__ void node_pre_kernel(const float* __restrict__ x,
                                const float* __restrict__ att_l,
                                const float* __restrict__ att_r,
                                float* __restrict__ al, float* __restrict__ ar,
                                float* __restrict__ deg, float* __restrict__ agg,
                                int N) {
    int wave = (int)((blockIdx.x * (unsigned)blockDim.x + threadIdx.x) >> 5);
    int lane = threadIdx.x & 31;
    if (wave >= N) return;
    const float4 xv = ((const float4*)(x + (size_t)wave * DF))[lane];
    const float4 lv = ((const float4*)att_l)[lane];
    const float4 rv = ((const float4*)att_r)[lane];
    float sl = xv.x*lv.x + xv.y*lv.y + xv.z*lv.z + xv.w*lv.w;
    float sr = xv.x*rv.x + xv.y*rv.y + xv.z*rv.z + xv.w*rv.w;
    #pragma unroll
    for (int off = 16; off > 0; off >>= 1) {
        sl += __shfl_xor(sl, off, 32);
        sr += __shfl_xor(sr, off, 32);
    }
    float4 z = {0.f, 0.f, 0.f, 0.f};
    ((float4*)(agg + (size_t)wave * DF))[lane] = z;
    if (lane == 0) { al[wave] = sl; ar[wave] = sr; deg[wave] = 1.0f; }
}

// ---------------------------------------------------------------------------
// Kernel 2: degree over edge targets, one thread per edge (native fp32 atomic).
// ---------------------------------------------------------------------------
__global__ void degree_kernel(const long long* __restrict__ ei,
                              float* __restrict__ deg, long long E) {
    long long e = blockIdx.x * (long long)blockDim.x + threadIdx.x;
    if (e >= E) return;
    long long c = ei[E + e];
    unsafeAtomicAdd(&deg[c], 1.0f);          // -> global_atomic_add_f32
}

// ---------------------------------------------------------------------------
// Kernel 3: dinv = rsqrt(deg); selfcoef = tanh(al+ar)*dinv^2 + EPS.
// ---------------------------------------------------------------------------
__global__ void node_coef_kernel(const float* __restrict__ deg,
                                 const float* __restrict__ al,
                                 const float* __restrict__ ar,
                                 float* __restrict__ dinv,
                                 float* __restrict__ selfcoef, int N) {
    int n = blockIdx.x * blockDim.x + threadIdx.x;
    if (n >= N) return;
    float d  = deg[n];
    float di = d > 0.f ? rsqrtf(d) : 0.f;
    dinv[n] = di;
    selfcoef[n] = tanhf(al[n] + ar[n]) * di * di + EPS;
}

// ---------------------------------------------------------------------------
// Kernel 4: one wave32 per edge: float4 gather of x[row] (L2-resident) and
// 4 native fp32 atomics per lane into agg[col].
// ---------------------------------------------------------------------------
__global__ void edge_scatter_kernel(const long long* __restrict__ ei,
                                    const float* __restrict__ x,
                                    const float* __restrict__ al,
                                    const float* __restrict__ ar,
                                    const float* __restrict__ dinv,
                                    float* __restrict__ agg, long long E) {
    long long wave = (blockIdx.x * (long long)blockDim.x + threadIdx.x) >> 5;
    int lane = threadIdx.x & 31;
    if (wave >= E) return;
    long long r = ei[wave];
    long long c = ei[E + wave];
    float coef = tanhf(al[r] + ar[c]) * dinv[r] * dinv[c];
    float4 xv = ((const float4*)(x + (size_t)r * DF))[lane];
    float* dst = agg + (size_t)c * DF + lane * 4;
    unsafeAtomicAdd(dst + 0, xv.x * coef);
    unsafeAtomicAdd(dst + 1, xv.y * coef);
    unsafeAtomicAdd(dst + 2, xv.z * coef);
    unsafeAtomicAdd(dst + 3, xv.w * coef);
}

// ---------------------------------------------------------------------------
// Kernel 4b: tiny one-shot transpose of weight [K x N] -> wT_g [N x K] in
// workspace, so the TDM can DMA it into the padded LDS layout directly.
// ---------------------------------------------------------------------------
__global__ void transpose_w_kernel(const float* __restrict__ w,
                                   float* __restrict__ wt) {
    int idx = blockIdx.x * blockDim.x + threadIdx.x;   // 0..16383
    if (idx >= DF * DF) return;
    int k = idx >> 7;
    int n = idx & (DF - 1);
    wt[n * DF + k] = w[idx];
}

// ---------------------------------------------------------------------------
// Kernel 5: fused finalize + GEMM with V_WMMA_F32_16X16X4_F32.
//   h[m][k] = agg[m][k] + selfcoef[m]*x[m][k]   (built in the A fragment)
//   out     = h @ W + bias
// Weight^T is brought into LDS by the Tensor Data Mover: a single
// tensor_load_to_lds whose descriptor enables padding (pad_interval=6 ->
// every 128 DWORDs insert pad_amount=3 -> 4 DWORDs), depositing a 128x128
// f32 tile with row stride 132 floats. That stride makes every wave32
// ds_load_b64 of a B fragment hit 64 distinct banks, and each B fragment is
// a single aligned b64 load (no repacking moves).
// One wave per 16-row tile; 8 column tiles => 8 v8f accumulators; 32 K-steps.
// A pair loads are software-pipelined one step ahead of the WMMA burst.
// ---------------------------------------------------------------------------
__global__ void gemm_wmma_kernel(const float* __restrict__ agg,
                                 const float* __restrict__ x,
                                 const float* __restrict__ selfcoef,
                                 const float* __restrict__ wt_g,  // W^T [N x K]
                                 const float* __restrict__ bias,
                                 float* __restrict__ out, int N) {
    __shared__ float wT[DF * WT_STRIDE];   // 67.6 KB of 320 KB WGP LDS

    if (threadIdx.x < 32) {
        // --- Tensor DMA descriptor (D#), ISA ch.8 ---
        unsigned lds_off = (unsigned)(uintptr_t)(&wT[0]);        // low 32 bits = LDS byte offset
        unsigned long long ga = (unsigned long long)(uintptr_t)wt_g;
        u32x4 g0;
        g0.x = 1u;                                               // count=1 (valid), user mode
        g0.y = lds_off;                                          // lds_addr
        g0.z = (unsigned)(ga & 0xffffffffull);                   // global_addr[31:0]
        g0.w = (unsigned)((ga >> 32) & 0x01ffffffull) | (2u << 30); // addr[56:32] | type=2
        i32x8 g1;
        g1[0] = (2 << 16)          // data_size = 4B
              | (1 << 20)          // pad_enable
              | (6 << 22)          // pad_interval -> every 128 DWORDs
              | (3 << 25);         // pad_amount -> 4 DWORDs => LDS row stride 132
        g1[1] = (DF & 0xffff) << 16;          // tensor_dim0 = 128 (lo16 at bits 63:48)
        g1[2] = (DF & 0xffff) << 16;          // tensor_dim1 = 128
        g1[3] = (DF & 0xffff) << 16;          // tile_dim0 = 128
        g1[4] = DF;                           // tile_dim1 = 128
        g1[5] = DF;                           // tensor_dim0_stride = 128
        g1[6] = ((DF * DF) & 0xffff) << 16;   // tensor_dim1_stride (lo16)
        g1[7] = 0;
        i32x4 z4 = {0, 0, 0, 0};              // groups 2/3 unused (2D tensor)
        i32x8 z8 = {0, 0, 0, 0, 0, 0, 0, 0};  // extra group (6-arg toolchain form)
        __builtin_amdgcn_tensor_load_to_lds(g0, g1, z4, z4, z8, 0);
        __builtin_amdgcn_s_wait_tensorcnt(0);
    }
    __syncthreads();

    const int lane = threadIdx.x & 31;
    const int wib  = threadIdx.x >> 5;
    const int tile = blockIdx.x * (blockDim.x >> 5) + wib;
    const int ntiles = (N + 15) >> 4;
    if (tile >= ntiles) return;               // wave-uniform: EXEC stays all-1s

    const int m0   = tile << 4;
    const int mrow = m0 + (lane & 15);
    const bool valid = (mrow < N);
    const int mrowc = valid ? mrow : (N - 1);
    const float mask = valid ? 1.f : 0.f;     // tail-row masking w/o divergence
    const int ksel = (lane >> 4) << 1;        // 0 for lanes 0-15, 2 for 16-31

    const float cself = selfcoef[mrowc];
    const float* aggRow = agg + (size_t)mrowc * DF;
    const float* xRow   = x   + (size_t)mrowc * DF;
    const float* wbase  = &wT[(lane & 15) * WT_STRIDE + ksel];

    v8f acc[8];
    #pragma unroll
    for (int ct = 0; ct < 8; ++ct) acc[ct] = (v8f){0,0,0,0,0,0,0,0};

    // software-pipelined A pair
    v2f ag = *(const v2f*)(aggRow + ksel);
    v2f ax = *(const v2f*)(xRow   + ksel);

    #pragma unroll 4
    for (int k0 = 0; k0 < DF; k0 += 4) {
        v2f a = (ag + cself * ax) * mask;
        if (k0 + 4 < DF) {                    // prefetch next k-step's pair
            ag = *(const v2f*)(aggRow + k0 + 4 + ksel);
            ax = *(const v2f*)(xRow   + k0 + 4 + ksel);
        }
        const float* wk = wbase + k0;
        #pragma unroll
        for (int ct = 0; ct < 8; ++ct) {
            v2f b = *(const v2f*)(wk + ct * 16 * WT_STRIDE);  // one ds_load_b64
            acc[ct] = __builtin_amdgcn_wmma_f32_16x16x4_f32(
                false, a, false, b, (short)0, acc[ct], false, false);
        }
    }

    const int rowHi = (lane >> 4) << 3;       // 0 or 8
    const int ncol  = lane & 15;
    #pragma unroll
    for (int ct = 0; ct < 8; ++ct) {
        const int ocol = (ct << 4) + ncol;
        const float bv = bias[ocol];
        #pragma unroll
        for (int r = 0; r < 8; ++r) {
            const int orow = m0 + rowHi + r;
            if (orow < N) out[(size_t)orow * DF + ocol] = acc[ct][r] + bv;
        }
    }
}

// ---------------------------------------------------------------------------
extern "C" void kernel_launch(void* const* d_in, const int* in_sizes, int n_in,
                              void* d_out, int out_size, void* d_ws, size_t ws_size,
                              hipStream_t stream) {
    const float*     x      = (const float*)d_in[0];
    const long long* ei     = (const long long*)d_in[1];   // int64 per reference
    const float*     att_l  = (const float*)d_in[2];
    const float*     att_r  = (const float*)d_in[3];
    const float*     weight = (const float*)d_in[4];
    const float*     bias   = (const float*)d_in[5];
    float*           out    = (float*)d_out;

    const int       N = in_sizes[0] / DF;
    const long long E = (long long)in_sizes[1] / 2;

    // workspace (floats): al | ar | deg | dinv | selfcoef | agg[N*DF] | wT[DF*DF]
    float* ws       = (float*)d_ws;
    float* al       = ws;
    float* ar       = al + N;
    float* deg      = ar + N;
    float* dinv     = deg + N;
    float* selfcoef = dinv + N;
    float* agg      = selfcoef + N;
    float* wt_g     = agg + (size_t)N * DF;

    {   // 8 waves (nodes) per 256-thread block
        int blocks = (N + 7) / 8;
        node_pre_kernel<<<blocks, 256, 0, stream>>>(x, att_l, att_r, al, ar, deg, agg, N);
    }
    {
        unsigned blocks = (unsigned)((E + 255) / 256);
        degree_kernel<<<blocks, 256, 0, stream>>>(ei, deg, E);
    }
    {
        int blocks = (N + 255) / 256;
        node_coef_kernel<<<blocks, 256, 0, stream>>>(deg, al, ar, dinv, selfcoef, N);
    }
    {   // 8 waves (edges) per 256-thread block
        unsigned blocks = (unsigned)((E + 7) / 8);
        edge_scatter_kernel<<<blocks, 256, 0, stream>>>(ei, x, al, ar, dinv, agg, E);
    }
    {
        transpose_w_kernel<<<(DF * DF + 255) / 256, 256, 0, stream>>>(weight, wt_g);
    }
    {   // 4 waves per block, one 16-row output tile per wave
        int ntiles = (N + 15) / 16;
        int blocks = (ntiles + 3) / 4;
        gemm_wmma_kernel<<<blocks, 128, 0, stream>>>(agg, x, selfcoef, wt_g, bias, out, N);
    }
}